// Attribute_Decoder_50723563766557
// MI455X (gfx1250) — compile-verified
//
#include <hip/hip_runtime.h>
#include <hip/hip_bf16.h>
#include <math.h>

#define N_NODES 100000
#define N_EDGES 1600000
#define E_TOT   (N_EDGES + N_NODES)   // self loops appended
#define NH      64
#define BN_EPS  1e-5f
#define NEG_SLOPE 0.2f

typedef float v2f __attribute__((ext_vector_type(2)));
typedef float v8f __attribute__((ext_vector_type(8)));

// ---------------------------------------------------------------------------
// init: m=-inf, s=0, agg=0   (must run every call; buffers are poisoned)
// ---------------------------------------------------------------------------
__global__ void k_init(float* __restrict__ m, float* __restrict__ s,
                       float* __restrict__ agg, int n, int total) {
    int i = blockIdx.x * blockDim.x + threadIdx.x;
    if (i < n) { m[i] = -INFINITY; s[i] = 0.0f; }
    if (i < total) agg[i] = 0.0f;
}

// ---------------------------------------------------------------------------
// GEMM: H[M x 64] = X[M x 64] * W[64 x 64], fp32 WMMA 16x16x4
// block = 256 threads = 8 waves; each wave owns a 16-row x 64-col stripe.
// ---------------------------------------------------------------------------
__global__ void k_gemm_wmma(const float* __restrict__ X,
                            const float* __restrict__ W,
                            float* __restrict__ H, int M) {
    __shared__ float sW[NH * NH];       // 16 KB
    __shared__ float sX[128 * NH];      // 32 KB

    const int tid  = threadIdx.x;
    const int wave = tid >> 5;
    const int lane = tid & 31;
    const int rowBlock = blockIdx.x * 128;

    // stage W (64x64) into LDS, coalesced float4
    for (int i = tid; i < NH * NH / 4; i += 256)
        ((float4*)sW)[i] = ((const float4*)W)[i];

    // stage 128x64 X tile into LDS, coalesced float4 (zero-fill past M)
    for (int i = tid; i < 128 * NH / 4; i += 256) {
        int r = rowBlock + (i >> 4);          // 16 float4 per row
        float4 v = make_float4(0.f, 0.f, 0.f, 0.f);
        if (r < M) v = ((const float4*)X)[(size_t)r * 16 + (i & 15)];
        ((float4*)sX)[i] = v;
    }
    __syncthreads();

    const int r0 = wave * 16;                  // row offset inside LDS tile
    if (rowBlock + r0 >= M) return;            // wave-uniform exit (EXEC stays full)

    const int lr = lane & 15;                  // 0..15
    const int hi = lane >> 4;                  // 0 or 1 (K half selector)

    v8f acc[4];
    for (int t = 0; t < 4; ++t) acc[t] = (v8f){0.f,0.f,0.f,0.f,0.f,0.f,0.f,0.f};

    for (int kb = 0; kb < NH; kb += 4) {
        // A frag (16x4 f32): lanes 0-15 hold K=kb,kb+1 of row lr; lanes 16-31 hold K=kb+2,kb+3
        v2f a;
        const float* arow = &sX[(r0 + lr) * NH + kb + 2 * hi];
        a.x = arow[0];
        a.y = arow[1];
        for (int t = 0; t < 4; ++t) {
            // B frag (4x16 f32): lanes 0-15 hold rows kb,kb+1; lanes 16-31 rows kb+2,kb+3; col = t*16+lr
            v2f b;
            const int col = t * 16 + lr;
            b.x = sW[(kb + 2 * hi + 0) * NH + col];
            b.y = sW[(kb + 2 * hi + 1) * NH + col];
            acc[t] = __builtin_amdgcn_wmma_f32_16x16x4_f32(
                false, a, false, b, (short)0, acc[t], false, false);
        }
    }

    // C/D layout: lane group picks M-half (v -> M=v or 8+v), N = t*16 + lr
    const int rbase = rowBlock + r0 + hi * 8;
    for (int t = 0; t < 4; ++t)
        for (int v = 0; v < 8; ++v)
            H[(size_t)(rbase + v) * NH + t * 16 + lr] = acc[t][v];
}

// ---------------------------------------------------------------------------
// per-node dots: asrc[i] = H[i,:].as ; adst[i] = H[i,:].ad  (one wave / node)
// ---------------------------------------------------------------------------
__global__ void k_node_dots(const float* __restrict__ H,
                            const float* __restrict__ aw_s,
                            const float* __restrict__ aw_d,
                            float* __restrict__ asrc, float* __restrict__ adst,
                            int n) {
    int node = (blockIdx.x * blockDim.x + threadIdx.x) >> 5;
    int lane = threadIdx.x & 31;
    if (node >= n) return;
    float2 hv = ((const float2*)(H + (size_t)node * NH))[lane];
    float2 s1 = ((const float2*)aw_s)[lane];
    float2 d1 = ((const float2*)aw_d)[lane];
    float ps = hv.x * s1.x + hv.y * s1.y;
    float pd = hv.x * d1.x + hv.y * d1.y;
    for (int off = 16; off > 0; off >>= 1) {
        ps += __shfl_down(ps, off, 32);
        pd += __shfl_down(pd, off, 32);
    }
    if (lane == 0) { asrc[node] = ps; adst[node] = pd; }
}

// float atomic max via sign-split integer atomics (init must be -inf)
__device__ __forceinline__ void atomicMaxF32(float* addr, float val) {
    if (val >= 0.0f) atomicMax((int*)addr, __float_as_int(val));
    else             atomicMin((unsigned int*)addr, __float_as_uint(val));
}

__device__ __forceinline__ void edge_ids(int e, const int* __restrict__ adj,
                                         int& src, int& dst) {
    if (e < N_EDGES) { src = adj[e]; dst = adj[N_EDGES + e]; }
    else             { src = dst = e - N_EDGES; }
}

// ---------------------------------------------------------------------------
// edge pass 1: e = LeakyReLU(asrc[src]+adst[dst]); segment max into m[dst]
// ---------------------------------------------------------------------------
__global__ void k_edge_logits(const int* __restrict__ adj,
                              const float* __restrict__ asrc,
                              const float* __restrict__ adst,
                              float* __restrict__ ebuf, float* __restrict__ m) {
    int e = blockIdx.x * blockDim.x + threadIdx.x;
    if (e >= E_TOT) return;
    int s_, d_;
    edge_ids(e, adj, s_, d_);
    float v = asrc[s_] + adst[d_];
    v = (v > 0.0f) ? v : NEG_SLOPE * v;
    ebuf[e] = v;
    atomicMaxF32(&m[d_], v);
}

// ---------------------------------------------------------------------------
// edge pass 2: w = exp(e - m[dst]) in place; segment sum into s[dst]
// ---------------------------------------------------------------------------
__global__ void k_edge_exp(const int* __restrict__ adj,
                           float* __restrict__ ebuf,
                           const float* __restrict__ m, float* __restrict__ s) {
    int e = blockIdx.x * blockDim.x + threadIdx.x;
    if (e >= E_TOT) return;
    int s_, d_;
    edge_ids(e, adj, s_, d_);
    float w = expf(ebuf[e] - m[d_]);
    ebuf[e] = w;
    atomicAdd(&s[d_], w);
}

// ---------------------------------------------------------------------------
// edge pass 3: agg[dst,:] += (w/s[dst]) * H[src,:]   (16 lanes per edge)
// ---------------------------------------------------------------------------
__global__ void k_edge_aggregate(const int* __restrict__ adj,
                                 const float* __restrict__ wbuf,
                                 const float* __restrict__ s,
                                 const float* __restrict__ H,
                                 float* __restrict__ agg) {
    long long gid = (long long)blockIdx.x * blockDim.x + threadIdx.x;
    int e  = (int)(gid >> 4);
    int c4 = (int)(gid & 15);          // float4 column group
    if (e >= E_TOT) return;
    int s_, d_;
    edge_ids(e, adj, s_, d_);
    float coeff = wbuf[e] / (s[d_] + 1e-16f);
    float4 hv = ((const float4*)(H + (size_t)s_ * NH))[c4];
    float* o = agg + (size_t)d_ * NH + c4 * 4;
    atomicAdd(o + 0, coeff * hv.x);
    atomicAdd(o + 1, coeff * hv.y);
    atomicAdd(o + 2, coeff * hv.z);
    atomicAdd(o + 3, coeff * hv.w);
}

// ---------------------------------------------------------------------------
// finalize layer 1: x = BN(ReLU(agg + b))  (in place -> becomes layer-2 input)
// ---------------------------------------------------------------------------
__global__ void k_finalize1(float* __restrict__ agg, const float* __restrict__ b,
                            const float* __restrict__ g, const float* __restrict__ beta,
                            const float* __restrict__ mean, const float* __restrict__ var,
                            int total) {
    int i = blockIdx.x * blockDim.x + threadIdx.x;
    if (i >= total) return;
    int c = i & (NH - 1);
    float v = agg[i] + b[c];
    v = fmaxf(v, 0.0f);
    v = (v - mean[c]) * rsqrtf(var[c] + BN_EPS) * g[c] + beta[c];
    agg[i] = v;
}

// finalize layer 2: out = ReLU(out + b)  (in place on d_out)
__global__ void k_finalize2(float* __restrict__ out, const float* __restrict__ b,
                            int total) {
    int i = blockIdx.x * blockDim.x + threadIdx.x;
    if (i >= total) return;
    out[i] = fmaxf(out[i] + b[i & (NH - 1)], 0.0f);
}

// ---------------------------------------------------------------------------
extern "C" void kernel_launch(void* const* d_in, const int* in_sizes, int n_in,
                              void* d_out, int out_size, void* d_ws, size_t ws_size,
                              hipStream_t stream) {
    const float* x     = (const float*)d_in[0];
    const int*   adj   = (const int*)  d_in[1];
    const float* W1    = (const float*)d_in[2];
    const float* as1   = (const float*)d_in[3];
    const float* ad1   = (const float*)d_in[4];
    const float* b1    = (const float*)d_in[5];
    const float* bn_g  = (const float*)d_in[6];
    const float* bn_b  = (const float*)d_in[7];
    const float* bn_m  = (const float*)d_in[8];
    const float* bn_v  = (const float*)d_in[9];
    const float* W2    = (const float*)d_in[10];
    const float* as2   = (const float*)d_in[11];
    const float* ad2   = (const float*)d_in[12];
    const float* b2    = (const float*)d_in[13];
    float* out = (float*)d_out;

    // workspace layout (floats)
    float* H    = (float*)d_ws;            // N*64
    float* agg1 = H    + (size_t)N_NODES * NH;   // N*64 (becomes layer-2 input)
    float* asrc = agg1 + (size_t)N_NODES * NH;   // N
    float* adst = asrc + N_NODES;                // N
    float* mbuf = adst + N_NODES;                // N
    float* sbuf = mbuf + N_NODES;                // N
    float* ebuf = sbuf + N_NODES;                // E_TOT

    const int TPB = 256;
    const int total    = N_NODES * NH;
    const int gTotal   = (total + TPB - 1) / TPB;            // 25000
    const int gGemm    = (N_NODES / 16 + 7) / 8;             // 782 (8 waves/block)
    const int gDots    = (N_NODES * 32 + TPB - 1) / TPB;     // 12500
    const int gEdge    = (E_TOT + TPB - 1) / TPB;            // 6641
    const long long aggThreads = (long long)E_TOT * 16;
    const int gAgg     = (int)((aggThreads + TPB - 1) / TPB);// 106250

    // ---- layer 1 ----
    k_init<<<gTotal, TPB, 0, stream>>>(mbuf, sbuf, agg1, N_NODES, total);
    k_gemm_wmma<<<gGemm, TPB, 0, stream>>>(x, W1, H, N_NODES);
    k_node_dots<<<gDots, TPB, 0, stream>>>(H, as1, ad1, asrc, adst, N_NODES);
    k_edge_logits<<<gEdge, TPB, 0, stream>>>(adj, asrc, adst, ebuf, mbuf);
    k_edge_exp<<<gEdge, TPB, 0, stream>>>(adj, ebuf, mbuf, sbuf);
    k_edge_aggregate<<<gAgg, TPB, 0, stream>>>(adj, ebuf, sbuf, H, agg1);
    k_finalize1<<<gTotal, TPB, 0, stream>>>(agg1, b1, bn_g, bn_b, bn_m, bn_v, total);

    // ---- layer 2 (accumulate straight into d_out) ----
    k_init<<<gTotal, TPB, 0, stream>>>(mbuf, sbuf, out, N_NODES, total);
    k_gemm_wmma<<<gGemm, TPB, 0, stream>>>(agg1, W2, H, N_NODES);
    k_node_dots<<<gDots, TPB, 0, stream>>>(H, as2, ad2, asrc, adst, N_NODES);
    k_edge_logits<<<gEdge, TPB, 0, stream>>>(adj, asrc, adst, ebuf, mbuf);
    k_edge_exp<<<gEdge, TPB, 0, stream>>>(adj, ebuf, mbuf, sbuf);
    k_edge_aggregate<<<gAgg, TPB, 0, stream>>>(adj, ebuf, sbuf, H, out);
    k_finalize2<<<gTotal, TPB, 0, stream>>>(out, b2, total);
}